// decoder_14585708937461
// MI455X (gfx1250) — compile-verified
//
#include <hip/hip_runtime.h>

typedef __attribute__((ext_vector_type(2))) float v2f;
typedef __attribute__((ext_vector_type(8))) float v8f;

#define BSZ 128
#define LSZ 512
#define EHD 256
#define DHD 256
#define YSZ 8

// ---------------------------------------------------------------------------
// f32 WMMA GEMM: C[M,N] = A[M,K] @ B[K,N] (+ bias[N] if non-null)
// B is supplied K-pair-packed: BtP[((k>>1)*N + n)*2 + (k&1)] = B[k][n], so a
// lane fetches its two K-values with one coalesced global_load_b64.
// Workgroup = 256 threads (8 waves), computes a 64(M) x 128(N) C block:
// wave w owns N-tile w, 4 M-sub-tiles -> 4 independent WMMA accumulate chains
// sharing each B fragment.  M % 64 == 0, N % 128 == 0, K % 32 == 0.
// ---------------------------------------------------------------------------
__global__ __launch_bounds__(256) void gemm_wmma_f32(
    const float* __restrict__ A, int lda,
    const float* __restrict__ BtP, int N, int K,
    const float* __restrict__ bias,
    float* __restrict__ C, int ldc)
{
    __shared__ float As[64][33];   // 64 rows x 32 K, padded -> conflict free
    const int tid  = threadIdx.x;
    const int wave = tid >> 5;
    const int lane = tid & 31;
    const int m0   = blockIdx.x * 64;
    const int n0   = blockIdx.y * 128 + wave * 16;
    const int ml   = lane & 15;   // M (A frag) / N (B frag) index within tile
    const int kh   = lane >> 4;   // K half-select

    v8f acc0 = {}, acc1 = {}, acc2 = {}, acc3 = {};
    for (int k0 = 0; k0 < K; k0 += 32) {
        // cooperative coalesced load of 64x32 A slab (8 floats per thread)
        int f = tid;
        #pragma unroll
        for (int r = 0; r < 8; ++r) {
            int m  = f >> 5;
            int kk = f & 31;
            As[m][kk] = A[(size_t)(m0 + m) * lda + (k0 + kk)];
            f += 256;
        }
        __syncthreads();
        #pragma unroll
        for (int kk = 0; kk < 32; kk += 4) {
            const int kb = kk + 2 * kh;            // even
            // one b64 load: K-pair (kb, kb+1) packed adjacently
            v2f b = *(const v2f*)(BtP +
                     ((size_t)((k0 + kb) >> 1) * N + (n0 + ml)) * 2);
            v2f a;
            a.x = As[ml][kb];      a.y = As[ml][kb + 1];
            acc0 = __builtin_amdgcn_wmma_f32_16x16x4_f32(false, a, false, b, (short)0, acc0, false, false);
            a.x = As[16 + ml][kb]; a.y = As[16 + ml][kb + 1];
            acc1 = __builtin_amdgcn_wmma_f32_16x16x4_f32(false, a, false, b, (short)0, acc1, false, false);
            a.x = As[32 + ml][kb]; a.y = As[32 + ml][kb + 1];
            acc2 = __builtin_amdgcn_wmma_f32_16x16x4_f32(false, a, false, b, (short)0, acc2, false, false);
            a.x = As[48 + ml][kb]; a.y = As[48 + ml][kb + 1];
            acc3 = __builtin_amdgcn_wmma_f32_16x16x4_f32(false, a, false, b, (short)0, acc3, false, false);
        }
        __syncthreads();
    }
    const int n = n0 + ml;
    const float bv = (bias != nullptr) ? bias[n] : 0.f;
    #pragma unroll
    for (int r = 0; r < 8; ++r) {
        int m = m0 + r + 8 * kh;
        C[(size_t)(m)      * ldc + n] = acc0[r] + bv;
        C[(size_t)(m + 16) * ldc + n] = acc1[r] + bv;
        C[(size_t)(m + 32) * ldc + n] = acc2[r] + bv;
        C[(size_t)(m + 48) * ldc + n] = acc3[r] + bv;
    }
}

// ---------------------------------------------------------------------------
// Fused attention for one batch row b:
//   s[l]  = w2 . tanh(u[b,l,:] + v[b,:]) + b2
//   attn  = softmax_l(s)
//   ctx[e]= sum_l attn[l] * X[b,l,e]
// One workgroup (512 threads = 16 waves) per b.  u is L2-resident (64 MB).
// ---------------------------------------------------------------------------
__global__ __launch_bounds__(512) void attn_fused(
    const float* __restrict__ u, const float* __restrict__ v,
    const float* __restrict__ w2, const float* __restrict__ b2,
    const float* __restrict__ X, float* __restrict__ context)
{
    __shared__ float vb[EHD], w2s[EHD], sc[LSZ], red[16], ctxp[2][EHD];
    const int b    = blockIdx.x;
    const int tid  = threadIdx.x;
    const int wave = tid >> 5;
    const int lane = tid & 31;

    if (tid < EHD) { vb[tid] = v[b * EHD + tid]; w2s[tid] = w2[tid]; }
    __syncthreads();

    const float bias2 = b2[0];
    // ---- scores: each wave handles 32 consecutive rows, lane-split dot ----
    for (int i = 0; i < 32; ++i) {
        int l = wave * 32 + i;
        const float* up = u + ((size_t)b * LSZ + l) * EHD;
        float acc = 0.f;
        #pragma unroll
        for (int k = 0; k < 8; ++k) {
            int e = lane + 32 * k;
            acc += tanhf(up[e] + vb[e]) * w2s[e];
        }
        #pragma unroll
        for (int off = 16; off >= 1; off >>= 1)
            acc += __shfl_xor(acc, off, 32);
        if (lane == 0) sc[l] = acc + bias2;
    }
    __syncthreads();

    // ---- softmax over L = 512 (one score per thread) ----
    float x = sc[tid];
    float m = x;
    #pragma unroll
    for (int off = 16; off >= 1; off >>= 1)
        m = fmaxf(m, __shfl_xor(m, off, 32));
    if (lane == 0) red[wave] = m;
    __syncthreads();
    if (tid == 0) {
        float mm = red[0];
        for (int i = 1; i < 16; ++i) mm = fmaxf(mm, red[i]);
        red[0] = mm;
    }
    __syncthreads();
    const float mx = red[0];
    __syncthreads();
    float e = __expf(x - mx);
    float s = e;
    #pragma unroll
    for (int off = 16; off >= 1; off >>= 1)
        s += __shfl_xor(s, off, 32);
    if (lane == 0) red[wave] = s;
    __syncthreads();
    if (tid == 0) {
        float ss = 0.f;
        for (int i = 0; i < 16; ++i) ss += red[i];
        red[0] = ss;
    }
    __syncthreads();
    sc[tid] = e * (1.f / red[0]);
    __syncthreads();

    // ---- context: coalesced over e, threads split L in halves ----
    const int half = tid >> 8;
    const int e2   = tid & 255;
    float a = 0.f;
    const float* xp = X + ((size_t)b * LSZ + half * 256) * EHD + e2;
    for (int l = 0; l < 256; ++l)
        a += sc[half * 256 + l] * xp[(size_t)l * EHD];
    ctxp[half][e2] = a;
    __syncthreads();
    if (tid < EHD) context[b * EHD + tid] = ctxp[0][tid] + ctxp[1][tid];
}

// ---------------------------------------------------------------------------
// y_tilde = [context | y_prev] @ W_fc.T + b_fc      -> [B, Y]
// ---------------------------------------------------------------------------
__global__ void ytilde_kernel(const float* __restrict__ ctx,
                              const float* __restrict__ yprev,
                              const float* __restrict__ Wfc,
                              const float* __restrict__ bfc,
                              float* __restrict__ yt)
{
    int idx = blockIdx.x * blockDim.x + threadIdx.x;
    if (idx >= BSZ * YSZ) return;
    int b = idx >> 3, y = idx & 7;
    const float* wr = Wfc + y * (EHD + YSZ);
    const float* cb = ctx + b * EHD;
    float acc = bfc[y];
    for (int e = 0; e < EHD; ++e) acc += cb[e] * wr[e];
    const float* yp = yprev + b * YSZ;
    #pragma unroll
    for (int j = 0; j < YSZ; ++j) acc += yp[j] * wr[EHD + j];
    yt[idx] = acc;
}

__device__ __forceinline__ float gate_pre(const float* __restrict__ grow,
                                          const float* __restrict__ Wih,
                                          const float* __restrict__ bih,
                                          const float* __restrict__ bhh,
                                          const float ytv[YSZ], int g)
{
    float s = grow[g] + bih[g] + bhh[g];
    const float* w = Wih + g * YSZ;
    #pragma unroll
    for (int j = 0; j < YSZ; ++j) s += ytv[j] * w[j];
    return s;
}

__device__ __forceinline__ float sigm(float x) { return 1.f / (1.f + __expf(-x)); }

// ---------------------------------------------------------------------------
// LSTM pointwise update.  gates holds h@W_hh.T; here we add biases and the
// small K=8 y_tilde@W_ih.T term, apply nonlinearities, update hc in place.
// hc layout: [B][512] = [h(256) | c(256)].
// ---------------------------------------------------------------------------
__global__ void lstm_update_kernel(const float* __restrict__ gates,
                                   const float* __restrict__ yt,
                                   const float* __restrict__ Wih,
                                   const float* __restrict__ bih,
                                   const float* __restrict__ bhh,
                                   float* __restrict__ hc)
{
    int idx = blockIdx.x * blockDim.x + threadIdx.x;
    if (idx >= BSZ * EHD) return;
    int b = idx >> 8, e = idx & 255;
    float ytv[YSZ];
    #pragma unroll
    for (int j = 0; j < YSZ; ++j) ytv[j] = yt[b * YSZ + j];
    const float* grow = gates + (size_t)b * 1024;
    float ig = sigm(gate_pre(grow, Wih, bih, bhh, ytv, e));
    float fg = sigm(gate_pre(grow, Wih, bih, bhh, ytv, 256 + e));
    float gg = tanhf(gate_pre(grow, Wih, bih, bhh, ytv, 512 + e));
    float og = sigm(gate_pre(grow, Wih, bih, bhh, ytv, 768 + e));
    float cold = hc[b * 512 + 256 + e];
    float cn = fg * cold + ig * gg;
    float hn = og * tanhf(cn);
    hc[b * 512 + e]       = hn;
    hc[b * 512 + 256 + e] = cn;
}

// ---------------------------------------------------------------------------
// y_pred = [h_new | context] @ W_fin.T + b_fin  -> out[:, t, :] and y_prev
// ---------------------------------------------------------------------------
__global__ void ypred_kernel(const float* __restrict__ hc,
                             const float* __restrict__ ctx,
                             const float* __restrict__ Wfin,
                             const float* __restrict__ bfin,
                             float* __restrict__ out,
                             float* __restrict__ yprev, int t, int T)
{
    int idx = blockIdx.x * blockDim.x + threadIdx.x;
    if (idx >= BSZ * YSZ) return;
    int b = idx >> 3, y = idx & 7;
    const float* wr = Wfin + y * (DHD + EHD);
    const float* hp = hc + b * 512;
    const float* cb = ctx + b * EHD;
    float acc = bfin[y];
    for (int e = 0; e < DHD; ++e) acc += hp[e] * wr[e];
    for (int e = 0; e < EHD; ++e) acc += cb[e] * wr[DHD + e];
    out[((size_t)b * T + t) * YSZ + y] = acc;
    yprev[idx] = acc;
}

// ---------------------------------------------------------------------------
// Weight prep: K-pair-packed transposed panels for WMMA B-operand b64 loads.
// Packed layout: P[((k>>1)*N + n)*2 + (k&1)] = Borig[k][n]
//   Wt1hc: Borig[k][n] = W_attn1[n][k]        (K=512, N=256)
//   Wt1x : Borig[k][n] = W_attn1[n][512 + k]  (K=256, N=256)
//   Wthh : Borig[k][g] = W_hh[g][k]           (K=256, N=1024)
// idx enumerates packed positions -> coalesced writes.
// ---------------------------------------------------------------------------
__global__ void prep_weights(const float* __restrict__ W1,
                             const float* __restrict__ Whh,
                             float* __restrict__ Wt1hc,
                             float* __restrict__ Wt1x,
                             float* __restrict__ Wthh)
{
    int idx = blockIdx.x * blockDim.x + threadIdx.x;
    const int n1 = 512 * EHD;
    const int n2 = n1 + 256 * EHD;
    const int n3 = n2 + 256 * 1024;
    if (idx < n1) {            // N = 256
        int n = (idx >> 1) & 255;
        int k = ((idx >> 1) >> 8) * 2 + (idx & 1);
        Wt1hc[idx] = W1[n * 768 + k];
    } else if (idx < n2) {     // N = 256
        int j = idx - n1;
        int n = (j >> 1) & 255;
        int k = ((j >> 1) >> 8) * 2 + (j & 1);
        Wt1x[j] = W1[n * 768 + 512 + k];
    } else if (idx < n3) {     // N = 1024
        int j = idx - n2;
        int g = (j >> 1) & 1023;
        int k = ((j >> 1) >> 10) * 2 + (j & 1);
        Wthh[j] = Whh[g * EHD + k];
    }
}

__global__ void init_state(float* __restrict__ hc, float* __restrict__ yprev,
                           const float* __restrict__ yhist)
{
    int idx = blockIdx.x * blockDim.x + threadIdx.x;
    if (idx < BSZ * 512) hc[idx] = 0.f;
    else if (idx < BSZ * 512 + BSZ * YSZ)
        yprev[idx - BSZ * 512] = yhist[idx - BSZ * 512];
}

// ---------------------------------------------------------------------------
extern "C" void kernel_launch(void* const* d_in, const int* in_sizes, int n_in,
                              void* d_out, int out_size, void* d_ws, size_t ws_size,
                              hipStream_t stream)
{
    const float* X     = (const float*)d_in[0];   // [B, L, EH]
    const float* yhist = (const float*)d_in[1];   // [B, Y]
    const float* W1    = (const float*)d_in[2];   // [EH, 768]
    const float* b1    = (const float*)d_in[3];   // [EH]
    const float* W2    = (const float*)d_in[4];   // [1, EH]
    const float* b2    = (const float*)d_in[5];   // [1]
    const float* Wih   = (const float*)d_in[6];   // [1024, Y]
    const float* Whh   = (const float*)d_in[7];   // [1024, EH]
    const float* bih   = (const float*)d_in[8];
    const float* bhh   = (const float*)d_in[9];
    const float* Wfc   = (const float*)d_in[10];  // [Y, EH+Y]
    const float* bfc   = (const float*)d_in[11];
    const float* Wfin  = (const float*)d_in[12];  // [Y, DH+EH]
    const float* bfin  = (const float*)d_in[13];
    // d_in[14] = fc_horizon (device scalar); derive T from out_size instead
    // (cannot sync-copy under graph capture).
    const int T = out_size / (BSZ * YSZ);

    float* ws = (float*)d_ws;
    size_t o = 0;
    float* u     = ws + o; o += (size_t)BSZ * LSZ * EHD; // 64 MB, L2-resident
    float* Wt1hc = ws + o; o += (size_t)512 * EHD;
    float* Wt1x  = ws + o; o += (size_t)256 * EHD;
    float* Wthh  = ws + o; o += (size_t)256 * 1024;
    float* v     = ws + o; o += (size_t)BSZ * EHD;
    float* ctx   = ws + o; o += (size_t)BSZ * EHD;
    float* yt    = ws + o; o += (size_t)BSZ * YSZ;
    float* yprev = ws + o; o += (size_t)BSZ * YSZ;
    float* hc    = ws + o; o += (size_t)BSZ * 512;
    float* gates = ws + o; o += (size_t)BSZ * 1024;
    if (ws_size < o * sizeof(float)) return;  // scratch too small

    {
        int total = 512 * EHD + 256 * EHD + 256 * 1024;
        prep_weights<<<(total + 255) / 256, 256, 0, stream>>>(W1, Whh, Wt1hc, Wt1x, Wthh);
    }
    {
        int total = BSZ * 512 + BSZ * YSZ;
        init_state<<<(total + 255) / 256, 256, 0, stream>>>(hc, yprev, yhist);
    }
    // One-time: u[b,l,:] = X[b,l,:] @ W1x.T + b_attn1   (time-invariant part)
    gemm_wmma_f32<<<dim3((BSZ * LSZ) / 64, EHD / 128), 256, 0, stream>>>(
        X, EHD, Wt1x, EHD, EHD, b1, u, EHD);

    for (int t = 0; t < T; ++t) {
        // v = [h|c] @ W1hc.T   (K = 512)
        gemm_wmma_f32<<<dim3(BSZ / 64, EHD / 128), 256, 0, stream>>>(
            hc, 512, Wt1hc, EHD, 512, nullptr, v, EHD);
        // scores -> softmax -> context, fused per batch row
        attn_fused<<<BSZ, 512, 0, stream>>>(u, v, W2, b2, X, ctx);
        // y_tilde
        ytilde_kernel<<<(BSZ * YSZ + 255) / 256, 256, 0, stream>>>(ctx, yprev, Wfc, bfc, yt);
        // gates_pre = h @ W_hh.T   (h = first half of hc, lda = 512)
        gemm_wmma_f32<<<dim3(BSZ / 64, 1024 / 128), 256, 0, stream>>>(
            hc, 512, Wthh, 1024, EHD, nullptr, gates, 1024);
        // LSTM pointwise update (adds biases + y_tilde@W_ih.T)
        lstm_update_kernel<<<(BSZ * EHD + 255) / 256, 256, 0, stream>>>(
            gates, yt, Wih, bih, bhh, hc);
        // y_pred -> out[:, t, :] and feedback y_prev
        ypred_kernel<<<(BSZ * YSZ + 255) / 256, 256, 0, stream>>>(
            hc, ctx, Wfin, bfin, (float*)d_out, yprev, t, T);
    }
}